// Sage_63015760167231
// MI455X (gfx1250) — compile-verified
//
#include <hip/hip_runtime.h>

#define NN      50000
#define DHID    128
#define BN_EPS  1e-5f

typedef __attribute__((ext_vector_type(2))) float v2f;
typedef __attribute__((ext_vector_type(8))) float v8f;

// ---------------------------------------------------------------- utilities
__global__ void k_zero(float* __restrict__ p, int n) {
  int i = blockIdx.x * blockDim.x + threadIdx.x;
  if (i < n) p[i] = 0.0f;
}

__global__ void k_degree(const int* __restrict__ dst, float* __restrict__ deg, int e) {
  int i = blockIdx.x * blockDim.x + threadIdx.x;
  if (i < e) atomicAdd(&deg[dst[i]], 1.0f);
}

__global__ void k_invdeg(float* __restrict__ deg, int n) {
  int i = blockIdx.x * blockDim.x + threadIdx.x;
  if (i < n) deg[i] = 1.0f / fmaxf(deg[i], 1.0f);
}

// ------------------------------------------------- edge scatter (mean numerator)
// one wave32 per edge: lane handles 4 contiguous floats (float4 load, 4 f32 atomics).
// src/dst wave-uniform -> scalarized; node rows are L2-resident (25.6MB << 192MB).
__global__ void k_scatter(const float* __restrict__ x, const int* __restrict__ src,
                          const int* __restrict__ dst, float* __restrict__ msg, int e) {
  int gid  = blockIdx.x * blockDim.x + threadIdx.x;
  int edge = gid >> 5;
  int lane = gid & 31;
  if (edge >= e) return;
  int s = src[edge];
  int d = dst[edge];
  const float4 v = *(const float4*)(x + (size_t)s * DHID + lane * 4);
  float* o = msg + (size_t)d * DHID + lane * 4;
  atomicAdd(o + 0, v.x);
  atomicAdd(o + 1, v.y);
  atomicAdd(o + 2, v.z);
  atomicAdd(o + 3, v.w);
}

// ------------------------------------------------- weight pre-pack into WMMA frag order
// packed[t][ks][lane] = v2f{ B[kk][col], B[kk+1][col] } with the f32 16x16x4 lane map
// (half = lane>>4 picks K+2/K+3). One b64 coalesced load per fragment in the GEMM.
__global__ void k_pack(const float* __restrict__ w, float* __restrict__ p, int D) {
  int idx   = blockIdx.x * blockDim.x + threadIdx.x;
  int total = (D / 16) * 32 * 32;             // tiles * ksteps * lanes
  if (idx >= total) return;
  int lane = idx & 31;
  int ks   = (idx >> 5) & 31;
  int t    = idx >> 10;
  int half = lane >> 4, l16 = lane & 15;
  int kk   = ks * 4 + half * 2;
  int col  = t * 16 + l16;
  p[2 * idx + 0] = w[kk * D + col];
  p[2 * idx + 1] = w[(kk + 1) * D + col];
}

// ------------------------------------------------- fused SAGE GEMM (WMMA f32)
// out[N,D] = (msg*invdeg) @ wl + xin @ wr + bias,  K = 128.
// One wave computes a full 16 x D row strip: A fragments are loaded ONCE per K-step
// and reused across all D/16 column tiles (8x less A traffic than tile-per-wave).
template <int D, bool STATS>
__global__ void k_sage_gemm(const float* __restrict__ msg, const float* __restrict__ invdeg,
                            const float* __restrict__ xin, const float* __restrict__ bias,
                            const v2f* __restrict__ pkl, const v2f* __restrict__ pkr,
                            float* __restrict__ out, float* __restrict__ colsum,
                            float* __restrict__ colsumsq) {
  constexpr int NT  = D / 16;
  const int lane    = threadIdx.x & 31;
  const int wave    = threadIdx.x >> 5;
  const int tileRow = blockIdx.x * 4 + wave;
  const int row0    = tileRow * 16;
  if (row0 >= NN) return;

  const int half = lane >> 4;   // 0: K+0/K+1 , 1: K+2/K+3  (ISA A f32 layout)
  const int l16  = lane & 15;
  const int rowA = row0 + l16;

  const float  inv  = invdeg[rowA];
  const float* mrow = msg + (size_t)rowA * DHID;
  const float* xrow = xin + (size_t)rowA * DHID;
  const v2f*   pl   = pkl + lane;
  const v2f*   pr   = pkr + lane;

  v8f acc[NT];
#pragma unroll
  for (int t = 0; t < NT; ++t) acc[t] = (v8f){};

#pragma unroll 2
  for (int ks = 0; ks < 32; ++ks) {
    const int kk = ks * 4 + half * 2;
    v2f am = {mrow[kk] * inv, mrow[kk + 1] * inv};   // mean branch (b64 + pk_mul)
    v2f ax = {xrow[kk], xrow[kk + 1]};               // self branch
#pragma unroll
    for (int t = 0; t < NT; ++t) {
      v2f bl = pl[(t * 32 + ks) * 32];               // coalesced b64 frag load
      v2f br = pr[(t * 32 + ks) * 32];
      acc[t] = __builtin_amdgcn_wmma_f32_16x16x4_f32(false, am, false, bl, (short)0,
                                                     acc[t], false, false);
      acc[t] = __builtin_amdgcn_wmma_f32_16x16x4_f32(false, ax, false, br, (short)0,
                                                     acc[t], false, false);
    }
  }

#pragma unroll
  for (int t = 0; t < NT; ++t) {
    const int   col = t * 16 + l16;
    const float b   = bias[col];
    float s = 0.0f, s2 = 0.0f;
#pragma unroll
    for (int i = 0; i < 8; ++i) {
      const int   m = half * 8 + i;          // C/D layout: lanes 16-31 hold M=8..15
      const float v = acc[t][i] + b;
      out[(size_t)(row0 + m) * D + col] = v;
      if (STATS) { s += v; s2 += v * v; }
    }
    if (STATS) {
      atomicAdd(&colsum[col], s);
      atomicAdd(&colsumsq[col], s2);
    }
  }
}

// ------------------------------------------------- BatchNorm + ReLU (in place)
__global__ void k_bn_relu(float* __restrict__ h, const float* __restrict__ colsum,
                          const float* __restrict__ colsumsq, const float* __restrict__ g,
                          const float* __restrict__ be, int n) {
  int i = blockIdx.x * blockDim.x + threadIdx.x;
  if (i >= n) return;
  const int   c   = i & (DHID - 1);
  const float mu  = colsum[c] * (1.0f / NN);
  const float var = colsumsq[c] * (1.0f / NN) - mu * mu;
  const float rs  = rsqrtf(var + BN_EPS);
  const float v   = (h[i] - mu) * rs * g[c] + be[c];
  h[i] = fmaxf(v, 0.0f);
}

// ---------------------------------------------------------------- launcher
extern "C" void kernel_launch(void* const* d_in, const int* in_sizes, int n_in,
                              void* d_out, int out_size, void* d_ws, size_t ws_size,
                              hipStream_t stream) {
  (void)n_in; (void)out_size; (void)ws_size;

  const float* x   = (const float*)d_in[0];
  const int*   ei  = (const int*)d_in[1];
  const int    E   = in_sizes[1] / 2;
  const int*   src = ei;
  const int*   dst = ei + E;
  const float *w1l = (const float*)d_in[2],  *w1r = (const float*)d_in[3];
  const float *b1  = (const float*)d_in[4],  *g1  = (const float*)d_in[5];
  const float *be1 = (const float*)d_in[6];
  const float *w2l = (const float*)d_in[7],  *w2r = (const float*)d_in[8];
  const float *b2  = (const float*)d_in[9],  *g2  = (const float*)d_in[10];
  const float *be2 = (const float*)d_in[11];
  const float *w3l = (const float*)d_in[12], *w3r = (const float*)d_in[13];
  const float *b3  = (const float*)d_in[14];

  // ws layout (f32): invdeg[N] | msg[N*128] | h1[N*128] | h2[N*128] | stats[256]
  //                 | pkl[16384] | pkr[16384]
  float* ws      = (float*)d_ws;
  float* invdeg  = ws;
  float* msg     = invdeg + NN;
  float* h1      = msg + (size_t)NN * DHID;
  float* h2      = h1 + (size_t)NN * DHID;
  float* stats   = h2 + (size_t)NN * DHID;
  float* colsum  = stats;
  float* colsmsq = stats + DHID;
  float* pkl     = stats + 2 * DHID;
  float* pkr     = pkl + 16384;

  const int TB       = 256;
  const int nFeat    = NN * DHID;
  const int gFeat    = (nFeat + TB - 1) / TB;
  const int gNode    = (NN + TB - 1) / TB;
  const int gEdge    = (E + TB - 1) / TB;
  const int gScatter = (E * 32 + TB - 1) / TB;
  const int rowTiles = (NN + 15) / 16;              // 3125
  dim3 gemmBlk(128);                                // 4 waves = 4 row strips
  dim3 gemmGrid((rowTiles + 3) / 4);                // 782
  const int gPack128 = (8 * 32 * 32 + TB - 1) / TB; // D=128 frag count
  const int gPack64  = (4 * 32 * 32 + TB - 1) / TB; // D=64

  // degree (shared by all layers), stored as 1/max(deg,1)
  k_zero<<<gNode, TB, 0, stream>>>(invdeg, NN);
  k_degree<<<gEdge, TB, 0, stream>>>(dst, invdeg, E);
  k_invdeg<<<gNode, TB, 0, stream>>>(invdeg, NN);

  // ---------------- layer 1: x -> h1
  k_pack<<<gPack128, TB, 0, stream>>>(w1l, pkl, DHID);
  k_pack<<<gPack128, TB, 0, stream>>>(w1r, pkr, DHID);
  k_zero<<<gFeat, TB, 0, stream>>>(msg, nFeat);
  k_zero<<<1, TB, 0, stream>>>(stats, 2 * DHID);
  k_scatter<<<gScatter, TB, 0, stream>>>(x, src, dst, msg, E);
  k_sage_gemm<DHID, true><<<gemmGrid, gemmBlk, 0, stream>>>(
      msg, invdeg, x, b1, (const v2f*)pkl, (const v2f*)pkr, h1, colsum, colsmsq);
  k_bn_relu<<<gFeat, TB, 0, stream>>>(h1, colsum, colsmsq, g1, be1, nFeat);

  // ---------------- layer 2: h1 -> h2
  k_pack<<<gPack128, TB, 0, stream>>>(w2l, pkl, DHID);
  k_pack<<<gPack128, TB, 0, stream>>>(w2r, pkr, DHID);
  k_zero<<<gFeat, TB, 0, stream>>>(msg, nFeat);
  k_zero<<<1, TB, 0, stream>>>(stats, 2 * DHID);
  k_scatter<<<gScatter, TB, 0, stream>>>(h1, src, dst, msg, E);
  k_sage_gemm<DHID, true><<<gemmGrid, gemmBlk, 0, stream>>>(
      msg, invdeg, h1, b2, (const v2f*)pkl, (const v2f*)pkr, h2, colsum, colsmsq);
  k_bn_relu<<<gFeat, TB, 0, stream>>>(h2, colsum, colsmsq, g2, be2, nFeat);

  // ---------------- layer 3: h2 -> d_out  (D=64, no BN/ReLU)
  k_pack<<<gPack64, TB, 0, stream>>>(w3l, pkl, 64);
  k_pack<<<gPack64, TB, 0, stream>>>(w3r, pkr, 64);
  k_zero<<<gFeat, TB, 0, stream>>>(msg, nFeat);
  k_scatter<<<gScatter, TB, 0, stream>>>(h2, src, dst, msg, E);
  k_sage_gemm<64, false><<<gemmGrid, gemmBlk, 0, stream>>>(
      msg, invdeg, h2, b3, (const v2f*)pkl, (const v2f*)pkr, (float*)d_out,
      nullptr, nullptr);
}